// FearFreeSOTA_77240691851789
// MI455X (gfx1250) — compile-verified
//
#include <hip/hip_runtime.h>
#include <hip/hip_bf16.h>
#include <math.h>

typedef __attribute__((ext_vector_type(2))) float v2f;
typedef __attribute__((ext_vector_type(8))) float v8f;

constexpr int FDIM  = 128;   // F_IN == F_OUT
constexpr int HEADS = 8;
constexpr int CH    = 16;
constexpr float SLOPE = 0.2f;

// ---- ordered-uint encoding for float atomic max --------------------------
__device__ __forceinline__ unsigned f2ord(float f) {
  unsigned u = __float_as_uint(f);
  return (u & 0x80000000u) ? ~u : (u | 0x80000000u);
}
__device__ __forceinline__ float ord2f(unsigned u) {
  return (u & 0x80000000u) ? __uint_as_float(u & 0x7FFFFFFFu) : __uint_as_float(~u);
}

// ---- Kernel 1: fused triple GEMM via V_WMMA_F32_16X16X4_F32 --------------
// Each block: 16 rows of raw = concat(x, time_feat) staged in LDS,
// 8 waves each own one 16-col stripe and run 3 WMMA accumulator chains.
__global__ void __launch_bounds__(256)
gat_gemm3(const float* __restrict__ x, const float* __restrict__ tf,
          const float* __restrict__ Wl, const float* __restrict__ Wr,
          const float* __restrict__ Ws,
          float* __restrict__ xl, float* __restrict__ xr,
          float* __restrict__ skip, int N, int DX) {
  __shared__ float tileA[16][FDIM + 4];
  const int row0 = blockIdx.x * 16;
  const int tid  = threadIdx.x;

  // build raw tile (126 cols from x, 2 cols from time_feat), zero-pad tail rows
  for (int i = 0; i < 8; ++i) {
    int e = i * 256 + tid;
    int r = e >> 7, c = e & 127;
    int row = row0 + r;
    float v = 0.f;
    if (row < N)
      v = (c < DX) ? x[(long long)row * DX + c]
                   : tf[(long long)row * 2 + (c - DX)];
    tileA[r][c] = v;
  }
  __syncthreads();

  const int wid  = tid >> 5;
  const int lane = tid & 31;
  const int m    = lane & 15;           // A-frag row (M)
  const int kh   = (lane >> 4) << 1;    // K sub-offset: lanes 16-31 hold K+2,K+3
  const int col  = wid * 16 + (lane & 15);

  v8f cl = {}, cr = {}, cs = {};
  for (int k0 = 0; k0 < FDIM; k0 += 4) {
    v2f a, bl, br, bs;
    a.x = tileA[m][k0 + kh];
    a.y = tileA[m][k0 + kh + 1];
    const float* wl0 = Wl + (long long)(k0 + kh) * FDIM + col;
    const float* wr0 = Wr + (long long)(k0 + kh) * FDIM + col;
    const float* ws0 = Ws + (long long)(k0 + kh) * FDIM + col;
    bl.x = wl0[0]; bl.y = wl0[FDIM];
    br.x = wr0[0]; br.y = wr0[FDIM];
    bs.x = ws0[0]; bs.y = ws0[FDIM];
    cl = __builtin_amdgcn_wmma_f32_16x16x4_f32(false, a, false, bl, (short)0, cl, false, false);
    cr = __builtin_amdgcn_wmma_f32_16x16x4_f32(false, a, false, br, (short)0, cr, false, false);
    cs = __builtin_amdgcn_wmma_f32_16x16x4_f32(false, a, false, bs, (short)0, cs, false, false);
  }

  // C/D layout: VGPR j -> M = j (lanes 0-15) / M = j+8 (lanes 16-31), N = lane%16
  const int rbase = row0 + ((lane >> 4) << 3);
  for (int j = 0; j < 8; ++j) {
    int row = rbase + j;
    if (row < N) {
      long long o = (long long)row * FDIM + col;
      xl[o] = cl[j]; xr[o] = cr[j]; skip[o] = cs[j];
    }
  }
}

// ---- shared edge-logit computation ---------------------------------------
__device__ __forceinline__ float edge_logit(const float* __restrict__ xl,
                                            const float* __restrict__ xr,
                                            const float* __restrict__ att,
                                            long long s, long long d, int h) {
  const float4* pl = (const float4*)(xl + s * FDIM + h * CH);
  const float4* pr = (const float4*)(xr + d * FDIM + h * CH);
  const float4* pa = (const float4*)(att + h * CH);
  float e = 0.f;
#pragma unroll
  for (int q = 0; q < 4; ++q) {
    float4 a = pl[q], b = pr[q], w = pa[q];
    float m0 = a.x + b.x, m1 = a.y + b.y, m2 = a.z + b.z, m3 = a.w + b.w;
    m0 = m0 > 0.f ? m0 : SLOPE * m0;
    m1 = m1 > 0.f ? m1 : SLOPE * m1;
    m2 = m2 > 0.f ? m2 : SLOPE * m2;
    m3 = m3 > 0.f ? m3 : SLOPE * m3;
    e += m0 * w.x + m1 * w.y + m2 * w.z + m3 * w.w;
  }
  return e;
}

__device__ __forceinline__ void edge_endpoints(const long long* __restrict__ ei,
                                               long long idx, long long E,
                                               long long& s, long long& d) {
  if (idx < E) { s = ei[idx]; d = ei[E + idx]; }
  else         { s = idx - E; d = idx - E; }        // self-loops
}

// ---- Kernel 2: per-(edge,head) logit + segment max (ordered-uint atomic) -
__global__ void __launch_bounds__(256)
gat_edge_max(const float* __restrict__ xl, const float* __restrict__ xr,
             const long long* __restrict__ ei, const float* __restrict__ att,
             unsigned* __restrict__ emax, long long E, long long N) {
  long long tid = (long long)blockIdx.x * blockDim.x + threadIdx.x;
  if (tid >= (E + N) * HEADS) return;
  long long idx = tid >> 3;
  int h = (int)(tid & 7);
  long long s, d;
  edge_endpoints(ei, idx, E, s, d);
  float e = edge_logit(xl, xr, att, s, d, h);
  atomicMax(&emax[d * HEADS + h], f2ord(e));
}

// ---- Kernel 3: exp, denom accumulate, unnormalized weighted aggregate ----
__global__ void __launch_bounds__(256)
gat_edge_acc(const float* __restrict__ xl, const float* __restrict__ xr,
             const long long* __restrict__ ei, const float* __restrict__ att,
             const unsigned* __restrict__ emax,
             float* __restrict__ denom, float* __restrict__ agg,
             long long E, long long N) {
  long long tid = (long long)blockIdx.x * blockDim.x + threadIdx.x;
  if (tid >= (E + N) * HEADS) return;
  long long idx = tid >> 3;
  int h = (int)(tid & 7);
  long long s, d;
  edge_endpoints(ei, idx, E, s, d);
  float e  = edge_logit(xl, xr, att, s, d, h);
  float mx = ord2f(emax[d * HEADS + h]);
  float ee = __expf(e - mx);
  atomicAdd(&denom[d * HEADS + h], ee);
  const float* pl = xl + s * FDIM + h * CH;
  float* pa = agg + d * FDIM + h * CH;
#pragma unroll
  for (int c = 0; c < CH; ++c)
    atomicAdd(&pa[c], ee * pl[c]);
}

// ---- Kernel 4: normalize, bias, skip, ELU, dot with Wo, sigmoid ----------
__global__ void __launch_bounds__(256)
gat_finalize(const float* __restrict__ agg, const float* __restrict__ denom,
             const float* __restrict__ skip, const float* __restrict__ bias_gat,
             const float* __restrict__ bs, const float* __restrict__ Wo,
             const float* __restrict__ bo, float* __restrict__ out, long long N) {
  int lane = threadIdx.x & 31;
  long long node = (long long)blockIdx.x * 8 + (threadIdx.x >> 5);
  if (node >= N) return;
  float sum = 0.f;
  int c0 = lane * 4;
#pragma unroll
  for (int j = 0; j < 4; ++j) {
    int c = c0 + j;
    int h = c >> 4;
    float g = agg[node * FDIM + c] / (denom[node * HEADS + h] + 1e-16f);
    float v = g + bias_gat[c] + skip[node * FDIM + c] + bs[c];
    v = v > 0.f ? v : (__expf(v) - 1.f);               // ELU (alpha=1)
    sum += v * Wo[c];
  }
  for (int off = 16; off > 0; off >>= 1)
    sum += __shfl_xor(sum, off, 32);
  if (lane == 0)
    out[node] = 1.f / (1.f + __expf(-(sum + bo[0])));
}

extern "C" void kernel_launch(void* const* d_in, const int* in_sizes, int n_in,
                              void* d_out, int out_size, void* d_ws, size_t ws_size,
                              hipStream_t stream) {
  const float*     x        = (const float*)d_in[0];
  const float*     tf       = (const float*)d_in[1];
  const long long* ei       = (const long long*)d_in[2];
  const float*     Wl       = (const float*)d_in[3];
  const float*     Wr       = (const float*)d_in[4];
  const float*     att      = (const float*)d_in[5];
  const float*     bias_gat = (const float*)d_in[6];
  const float*     Ws       = (const float*)d_in[7];
  const float*     bs       = (const float*)d_in[8];
  const float*     Wo       = (const float*)d_in[9];
  const float*     bo       = (const float*)d_in[10];

  const int DX = 126;
  const long long N = in_sizes[0] / DX;
  const long long E = in_sizes[2] / 2;

  // workspace layout (fp32 unless noted)
  float*    ws_f  = (float*)d_ws;
  float*    xl    = ws_f;                    // N*128
  float*    xr    = xl + N * FDIM;           // N*128
  float*    skip  = xr + N * FDIM;           // N*128
  float*    agg   = skip + N * FDIM;         // N*128  (zeroed)
  unsigned* emax  = (unsigned*)(agg + N * FDIM);   // N*8 (zeroed; ordered enc)
  float*    denom = (float*)(emax + N * HEADS);    // N*8  (zeroed)

  // zero agg + emax + denom in one contiguous memset (graph-capturable)
  hipMemsetAsync(agg, 0, (size_t)N * (FDIM + 2 * HEADS) * sizeof(float), stream);

  int gemmBlocks = (int)((N + 15) / 16);
  gat_gemm3<<<gemmBlocks, 256, 0, stream>>>(x, tf, Wl, Wr, Ws, xl, xr, skip,
                                            (int)N, DX);

  long long totalEH = (E + N) * HEADS;
  int edgeBlocks = (int)((totalEH + 255) / 256);
  gat_edge_max<<<edgeBlocks, 256, 0, stream>>>(xl, xr, ei, att, emax, E, N);
  gat_edge_acc<<<edgeBlocks, 256, 0, stream>>>(xl, xr, ei, att, emax,
                                               denom, agg, E, N);

  int finBlocks = (int)((N + 7) / 8);
  gat_finalize<<<finBlocks, 256, 0, stream>>>(agg, denom, skip, bias_gat, bs,
                                              Wo, bo, (float*)d_out, N);
}